// HeterogeneousSkipGram_32530082300267
// MI455X (gfx1250) — compile-verified
//
#include <hip/hip_runtime.h>

typedef __attribute__((ext_vector_type(2))) float v2f;
typedef __attribute__((ext_vector_type(4))) float v4f;
typedef __attribute__((ext_vector_type(8))) float v8f;

#define EMB   128
#define KNEG  20
#define BATCH 16384
#define STR   132   // padded LDS row stride (floats): bank-conflict-free, 16B-aligned
#define WPB   2     // waves per block

__device__ __forceinline__ float log_sigmoid(float x) {
    // log(sigmoid(x)) = min(x,0) - log1p(exp(-|x|))  (numerically stable)
    return fminf(x, 0.0f) - log1pf(__expf(-fabsf(x)));
}

__global__ void sg_zero_out(float* out) { out[0] = 0.0f; }

// One wave handles a tile of 16 batch rows.
// Phase 1: coalesced global_load_b128 staging (each 512B embedding row read
//          exactly once, full cachelines); 20 negative rows reduced in
//          registers, only the sum hits LDS.
// Phase 2: ds_load_b64 in WMMA layout + v_wmma_f32_16x16x4_f32 diagonal dots.
__global__ __launch_bounds__(32 * WPB) void HeterogeneousSkipGram_kernel(
    const int*   __restrict__ u,
    const int*   __restrict__ v,
    const int*   __restrict__ neg_v,
    const float* __restrict__ u_table,
    const float* __restrict__ v_table,
    float*       __restrict__ out)
{
    __shared__ float lds[WPB * 3 * 16 * STR];

    const int lane = threadIdx.x & 31;
    const int wave = threadIdx.x >> 5;
    const int tile = blockIdx.x * WPB + wave;      // 1024 tiles of 16 rows

    float* us = lds + wave * (3 * 16 * STR);
    float* vs = us + 16 * STR;
    float* ns = vs + 16 * STR;

    const int c4 = lane * 4;                       // float offset of this lane's 16B chunk

    // ---------------- Phase 1: coalesced staging ----------------
    for (int m = 0; m < 16; ++m) {
        const int r = tile * 16 + m;               // wave-uniform row index

        v4f a = *(const v4f*)(u_table + (long long)u[r] * EMB + c4);
        *(v4f*)(us + m * STR + c4) = a;

        v4f b = *(const v4f*)(v_table + (long long)v[r] * EMB + c4);
        *(v4f*)(vs + m * STR + c4) = b;

        v4f acc = {0.0f, 0.0f, 0.0f, 0.0f};
#pragma unroll
        for (int j = 0; j < KNEG; ++j) {
            v4f t = *(const v4f*)(v_table + (long long)neg_v[r * KNEG + j] * EMB + c4);
            acc += t;
        }
        *(v4f*)(ns + m * STR + c4) = acc;
    }

    __syncthreads();

    // ---------------- Phase 2: WMMA diagonal dots ----------------
    // f32 WMMA 16x16x4 (wave32): A lanes 0-15 hold M=lane,K={k,k+1};
    // lanes 16-31 hold M=lane-16,K={k+2,k+3}; B mirrors with N.
    const int mrow = (lane & 15) * STR;
    const int hi   = (lane >> 4) << 1;             // 0 or 2

    v8f cpos = {};
    v8f cneg = {};
#pragma unroll 4
    for (int k = 0; k < EMB; k += 4) {
        const int off = k + hi;
        v2f a  = *(const v2f*)(us + mrow + off);
        v2f bp = *(const v2f*)(vs + mrow + off);
        v2f bn = *(const v2f*)(ns + mrow + off);
        cpos = __builtin_amdgcn_wmma_f32_16x16x4_f32(false, a, false, bp,
                                                     (short)0, cpos, false, false);
        cneg = __builtin_amdgcn_wmma_f32_16x16x4_f32(false, a, false, bn,
                                                     (short)0, cneg, false, false);
    }

    // Diagonal D[n][n]: (lane=n, vgpr=n) for n<8; (lane=n+16, vgpr=n-8) for n>=8.
    const int sel = (lane < 8) ? lane : ((lane >= 24) ? (lane - 24) : -1);
    float dp = 0.0f, dn = 0.0f;
#pragma unroll
    for (int i = 0; i < 8; ++i) {
        dp = (sel == i) ? cpos[i] : dp;
        dn = (sel == i) ? cneg[i] : dn;
    }

    float contrib = 0.0f;
    if (sel >= 0) {
        float s  = fminf(fmaxf(dp, -10.0f), 10.0f);
        float nsc = fminf(fmaxf(dn, -10.0f), 10.0f);
        contrib = -(log_sigmoid(s) + log_sigmoid(-nsc)) * (1.0f / (float)BATCH);
    }

    // wave32 butterfly reduction
#pragma unroll
    for (int o = 16; o > 0; o >>= 1)
        contrib += __shfl_xor(contrib, o, 32);

    if (lane == 0)
        atomicAdd(out, contrib);
}

extern "C" void kernel_launch(void* const* d_in, const int* in_sizes, int n_in,
                              void* d_out, int out_size, void* d_ws, size_t ws_size,
                              hipStream_t stream) {
    (void)in_sizes; (void)n_in; (void)out_size; (void)d_ws; (void)ws_size;
    const int*   u       = (const int*)d_in[0];
    const int*   v       = (const int*)d_in[1];
    const int*   neg_v   = (const int*)d_in[2];
    const float* u_table = (const float*)d_in[3];
    const float* v_table = (const float*)d_in[4];
    float*       out     = (float*)d_out;

    hipLaunchKernelGGL(sg_zero_out, dim3(1), dim3(1), 0, stream, out);
    // 1024 tiles of 16 rows; WPB waves per block
    hipLaunchKernelGGL(HeterogeneousSkipGram_kernel,
                       dim3(BATCH / 16 / WPB), dim3(32 * WPB), 0, stream,
                       u, v, neg_v, u_table, v_table, out);
}